// Axis_Horizontal_Attention_block_9706626089298
// MI455X (gfx1250) — compile-verified
//
#include <hip/hip_runtime.h>
#include <hip/hip_bf16.h>
#include <cstddef>
#include <cstdint>

typedef __attribute__((ext_vector_type(2))) float v2f;
typedef __attribute__((ext_vector_type(8))) float v8f;

namespace {
constexpr int    BATCH = 16;
constexpr int    CDIM  = 256;
constexpr int    HDIM  = 128;
constexpr int    WDIM  = 128;
constexpr int    HW    = HDIM * WDIM;           // 16384
constexpr size_t CHW   = (size_t)CDIM * HW;     // 4194304 elements per batch
constexpr int    MROW  = CDIM * WDIM;           // 32768 = q row length = out1 row count

constexpr size_t SZ_BIG  = (size_t)BATCH * CHW; // 67108864
constexpr size_t OUT_OUT = 0;
constexpr size_t OUT_Q   = SZ_BIG;
constexpr size_t OUT_K   = 2 * SZ_BIG;
constexpr size_t OUT_V   = 3 * SZ_BIG;
constexpr size_t OUT_G   = 4 * SZ_BIG;
constexpr size_t OUT_A   = OUT_G + 1;

__device__ inline v8f wmma_f32_16x16x4(v2f a, v2f b, v8f c) {
  // D = A(16x4 f32) * B(4x16 f32) + C(16x16 f32), exact fp32 matrix pipe
  return __builtin_amdgcn_wmma_f32_16x16x4_f32(false, a, false, b, (short)0, c,
                                               false, false);
}

// CDNA5 async copy engine: global -> LDS, 16B per lane, tracked by ASYNCcnt.
// VDST VGPR carries the per-lane LDS byte address (low 32 bits of the generic
// shared pointer), VADDR pair carries the 64-bit global address (GV mode).
__device__ inline void async_copy_b128(void* lds_dst, const void* gsrc) {
  uint32_t loff = (uint32_t)(uintptr_t)lds_dst;
  asm volatile("global_load_async_to_lds_b128 %0, %1, off"
               :
               : "v"(loff), "v"(gsrc)
               : "memory");
}
__device__ inline void async_wait0() {
  asm volatile("s_wait_asynccnt 0" ::: "memory");
}
}  // namespace

// ---------------------------------------------------------------------------
// Kernel 1: q/k/v = relu(W @ x) ; q stored natural, k/v stored transposed
// grid.x = b*128 + pixel_tile(128px) ; grid.y = proj*2 + o_half ; 256 thr
// ---------------------------------------------------------------------------
__global__ __launch_bounds__(256) void proj_kernel(
    const float* __restrict__ x, const float* __restrict__ Wq,
    const float* __restrict__ Wk, const float* __restrict__ Wv,
    float* __restrict__ out) {
  extern __shared__ float ldsW[];  // 128 x 256 fp32 = 128 KB (of 320 KB LDS)
  const int tid  = threadIdx.x;
  const int wv   = tid >> 5;
  const int lane = tid & 31;
  const int half = lane >> 4;
  const int lq   = lane & 15;

  const int b     = blockIdx.x >> 7;
  const int ptile = blockIdx.x & 127;
  const int proj  = blockIdx.y >> 1;
  const int obase = (blockIdx.y & 1) * 128;

  const float* __restrict__ Wm = (proj == 0) ? Wq : (proj == 1) ? Wk : Wv;

  {  // stage 128 output-channel rows of W straight into LDS (async engine)
    const float* src = Wm + (size_t)obase * CDIM;
    for (int i = tid; i < (128 * CDIM) / 4; i += 256)
      async_copy_b128(&ldsW[i * 4], &src[i * 4]);
    async_wait0();
  }
  __syncthreads();

  const int p = ptile * 128 + wv * 16 + lq;  // this lane's pixel column (N)
  const float* __restrict__ xb = x + (size_t)b * CHW;

  v8f acc[8] = {};
  for (int c0 = 0; c0 < CDIM; c0 += 4) {
    const int cA = c0 + 2 * half;
    v2f bv;
    bv.x = xb[(size_t)cA * HW + p];
    bv.y = xb[(size_t)(cA + 1) * HW + p];
    if ((c0 & 31) == 0 && c0 + 32 < CDIM)
      __builtin_prefetch(&xb[(size_t)(cA + 32) * HW + p], 0, 1);
#pragma unroll
    for (int t = 0; t < 8; ++t) {  // 8 M-tiles: o = obase + t*16 ...
      v2f a = *(const v2f*)&ldsW[(size_t)(t * 16 + lq) * CDIM + cA];
      acc[t] = wmma_f32_16x16x4(a, bv, acc[t]);
    }
  }

  const size_t rbase = OUT_Q + (size_t)proj * SZ_BIG + (size_t)b * CHW;
#pragma unroll
  for (int t = 0; t < 8; ++t) {
#pragma unroll
    for (int r = 0; r < 8; ++r) {
      float vv = acc[t][r];
      vv = vv > 0.0f ? vv : 0.0f;  // ReLU
      const int o = obase + t * 16 + r + 8 * half;
      if (proj == 0) {
        out[rbase + (size_t)o * HW + p] = vv;  // q: natural [B,C,H,W] flat
      } else {
        // k/v transposed to [B, C*W, H]: m=(o&1)*HW+p, column = o>>1
        out[rbase + (size_t)((o & 1) * HW + p) * HDIM + (o >> 1)] = vv;
      }
    }
  }
}

// ---------------------------------------------------------------------------
// Kernel 2: scores[b,i,j] = sum_m q[b,i,m]*k[b,j,m]  (K = 32768)
// grid.x = b*8 + i_tile ; 8 waves = 8 j_tiles ; result into attn region
// ---------------------------------------------------------------------------
__global__ __launch_bounds__(256) void scores_kernel(float* __restrict__ out) {
  const int tid = threadIdx.x;
  const int wv = tid >> 5, lane = tid & 31, half = lane >> 4, lq = lane & 15;
  const int b  = blockIdx.x >> 3;
  const int i0 = (blockIdx.x & 7) * 16;
  const int j0 = wv * 16;

  const float* __restrict__ qb = out + OUT_Q + (size_t)b * CHW;
  const float* __restrict__ kb = out + OUT_K + (size_t)b * CHW;

  v8f acc = {};
  const size_t qrow = (size_t)(i0 + lq) * MROW + 2 * half;
  for (int m0 = 0; m0 < MROW; m0 += 4) {
    v2f a = *(const v2f*)&qb[qrow + m0];
    const int mB = m0 + 2 * half;
    v2f bv;
    bv.x = kb[(size_t)mB * HDIM + j0 + lq];
    bv.y = kb[(size_t)(mB + 1) * HDIM + j0 + lq];
    if ((m0 & 255) == 0 && m0 + 256 < MROW)
      __builtin_prefetch(&qb[qrow + m0 + 256], 0, 1);
    acc = wmma_f32_16x16x4(a, bv, acc);
  }
  float* __restrict__ sb = out + OUT_A + (size_t)b * (HDIM * HDIM);
#pragma unroll
  for (int r = 0; r < 8; ++r)
    sb[(size_t)(i0 + r + 8 * half) * HDIM + j0 + lq] = acc[r];
}

// ---------------------------------------------------------------------------
// Kernel 3: softmax over axis=1 (the i rows), per column j, in place.
// Columns are the fast dim -> thread-per-column is coalesced.
// ---------------------------------------------------------------------------
__global__ __launch_bounds__(128) void softmax_kernel(
    const float* __restrict__ gamma, float* __restrict__ out) {
  const int b = blockIdx.x, j = threadIdx.x;
  float* __restrict__ sb = out + OUT_A + (size_t)b * (HDIM * HDIM);
  float mx = -3.402823466e38f;
  for (int i = 0; i < HDIM; ++i) mx = fmaxf(mx, sb[i * HDIM + j]);
  float sum = 0.0f;
  for (int i = 0; i < HDIM; ++i) sum += __expf(sb[i * HDIM + j] - mx);
  const float inv = 1.0f / sum;
  for (int i = 0; i < HDIM; ++i)
    sb[i * HDIM + j] = __expf(sb[i * HDIM + j] - mx) * inv;
  if (b == 0 && j == 0) out[OUT_G] = gamma[0];  // gamma passthrough output
}

// ---------------------------------------------------------------------------
// Kernel 4: out1[b,m,j] = gamma * sum_i v_t[b,m,i]*attn[b,i,j] -> ws
// (natural [B,M,H] layout of out1 == [B,C,H,W] flat layout of out_h)
// grid.x = b*256 + m_block(8 tiles) ; attn matrix (64 KB) staged via async
// ---------------------------------------------------------------------------
__global__ __launch_bounds__(256) void out1_kernel(
    const float* __restrict__ gamma, const float* __restrict__ out,
    float* __restrict__ ws) {
  extern __shared__ float latt[];  // 128 x 128 attn (64 KB)
  const int tid = threadIdx.x;
  const int wv = tid >> 5, lane = tid & 31, half = lane >> 4, lq = lane & 15;
  const int b  = blockIdx.x >> 8;
  const int mt = blockIdx.x & 255;

  {
    const float* src = out + OUT_A + (size_t)b * (HDIM * HDIM);
    for (int i = tid; i < (HDIM * HDIM) / 4; i += 256)
      async_copy_b128(&latt[i * 4], &src[i * 4]);
    async_wait0();
  }
  __syncthreads();

  const int m0 = (mt * 8 + wv) * 16;
  const float* __restrict__ vb = out + OUT_V + (size_t)b * CHW;
  v8f acc[8] = {};
  const size_t vrow = (size_t)(m0 + lq) * HDIM + 2 * half;
  for (int i0 = 0; i0 < HDIM; i0 += 4) {
    v2f a = *(const v2f*)&vb[vrow + i0];
    const int iB = i0 + 2 * half;
#pragma unroll
    for (int jt = 0; jt < 8; ++jt) {
      v2f bv;
      bv.x = latt[iB * HDIM + jt * 16 + lq];
      bv.y = latt[(iB + 1) * HDIM + jt * 16 + lq];
      acc[jt] = wmma_f32_16x16x4(a, bv, acc[jt]);
    }
  }
  const float g = gamma[0];
  float* __restrict__ wb = ws + (size_t)b * CHW;
#pragma unroll
  for (int jt = 0; jt < 8; ++jt)
#pragma unroll
    for (int r = 0; r < 8; ++r)
      wb[(size_t)(m0 + r + 8 * half) * HDIM + jt * 16 + lq] = acc[jt][r] * g;
}

// ---------------------------------------------------------------------------
// Kernel 5: out_h = Wup @ ws + b_up  (same shape as proj GEMM, + bias)
// ---------------------------------------------------------------------------
__global__ __launch_bounds__(256) void up_kernel(
    const float* __restrict__ hsrc, const float* __restrict__ Wup,
    const float* __restrict__ bup, float* __restrict__ out) {
  extern __shared__ float ldsW[];  // 128 x 256 fp32
  const int tid  = threadIdx.x;
  const int wv   = tid >> 5;
  const int lane = tid & 31;
  const int half = lane >> 4;
  const int lq   = lane & 15;

  const int b     = blockIdx.x >> 7;
  const int ptile = blockIdx.x & 127;
  const int obase = blockIdx.y * 128;

  {
    const float* src = Wup + (size_t)obase * CDIM;
    for (int i = tid; i < (128 * CDIM) / 4; i += 256)
      async_copy_b128(&ldsW[i * 4], &src[i * 4]);
    async_wait0();
  }
  __syncthreads();

  const int p = ptile * 128 + wv * 16 + lq;
  const float* __restrict__ hb = hsrc + (size_t)b * CHW;

  v8f acc[8] = {};
  for (int c0 = 0; c0 < CDIM; c0 += 4) {
    const int cA = c0 + 2 * half;
    v2f bv;
    bv.x = hb[(size_t)cA * HW + p];
    bv.y = hb[(size_t)(cA + 1) * HW + p];
    if ((c0 & 31) == 0 && c0 + 32 < CDIM)
      __builtin_prefetch(&hb[(size_t)(cA + 32) * HW + p], 0, 1);
#pragma unroll
    for (int t = 0; t < 8; ++t) {
      v2f a = *(const v2f*)&ldsW[(size_t)(t * 16 + lq) * CDIM + cA];
      acc[t] = wmma_f32_16x16x4(a, bv, acc[t]);
    }
  }

  const size_t rbase = OUT_OUT + (size_t)b * CHW;
#pragma unroll
  for (int t = 0; t < 8; ++t) {
#pragma unroll
    for (int r = 0; r < 8; ++r) {
      const int o = obase + t * 16 + r + 8 * half;
      out[rbase + (size_t)o * HW + p] = acc[t][r] + bup[o];
    }
  }
}

// ---------------------------------------------------------------------------
extern "C" void kernel_launch(void* const* d_in, const int* in_sizes, int n_in,
                              void* d_out, int out_size, void* d_ws,
                              size_t ws_size, hipStream_t stream) {
  const float* x     = (const float*)d_in[0];
  const float* Wq    = (const float*)d_in[1];
  const float* Wk    = (const float*)d_in[2];
  const float* Wv    = (const float*)d_in[3];
  const float* gamma = (const float*)d_in[4];
  const float* Wup   = (const float*)d_in[5];
  const float* bup   = (const float*)d_in[6];
  float* out = (float*)d_out;
  float* ws  = (float*)d_ws;  // needs B*C*H*W*4 = 256 MB scratch for out1

  proj_kernel<<<dim3(BATCH * 128, 6), 256, 128 * CDIM * sizeof(float), stream>>>(
      x, Wq, Wk, Wv, out);
  scores_kernel<<<dim3(BATCH * 8), 256, 0, stream>>>(out);
  softmax_kernel<<<dim3(BATCH), 128, 0, stream>>>(gamma, out);
  out1_kernel<<<dim3(BATCH * 256), 256, HDIM * HDIM * sizeof(float), stream>>>(
      gamma, out, ws);
  up_kernel<<<dim3(BATCH * 128, 2), 256, 128 * CDIM * sizeof(float), stream>>>(
      ws, Wup, bup, out);
}